// BackgroundNoiseLayer_36155034697743
// MI455X (gfx1250) — compile-verified
//
#include <hip/hip_runtime.h>

// Problem constants (match reference)
#define N_NEURONS 50000
#define N_BKG     100
#define N_RECEPT  5
#define N_EDGES   500000
#define SEQ_T     250
#define N_COLS    (N_NEURONS * N_RECEPT)   // 250000
#define M_PAD     256                      // SEQ_T padded to 16*16
#define W_ELEMS   (N_COLS * N_BKG)         // 25,000,000 floats
#define SPAD_OFF  W_ELEMS                  // padded spikes start (floats)
#define SPAD_ELEMS (M_PAD * N_BKG)         // 25,600 floats
#define TOT_INIT  (W_ELEMS + SPAD_ELEMS)

typedef __attribute__((ext_vector_type(2))) float v2f;
typedef __attribute__((ext_vector_type(8))) float v8f;

// ---------------------------------------------------------------------------
// Kernel 1: zero the dense Wt buffer and build a zero-padded copy of spikes.
// ws float layout: [0, W_ELEMS)            Wt[col][k], col = n*5 + r
//                  [W_ELEMS, +M_PAD*100)   spikes padded to 256 rows
// ---------------------------------------------------------------------------
__global__ void init_ws(float* __restrict__ ws, const float* __restrict__ spikes) {
    int stride = gridDim.x * blockDim.x;
    for (int i = blockIdx.x * blockDim.x + threadIdx.x; i < TOT_INIT; i += stride) {
        if (i < W_ELEMS) {
            ws[i] = 0.0f;
        } else {
            int j = i - SPAD_OFF;
            ws[i] = (j < SEQ_T * N_BKG) ? spikes[j] : 0.0f;
        }
    }
}

// ---------------------------------------------------------------------------
// Kernel 2: scatter-add edges into Wt.
// Wt[(n*5 + r)*100 + k] += weights[e] * tau_syn[e*5 + r]
// ---------------------------------------------------------------------------
__global__ void scatter_edges(const float* __restrict__ weights,
                              const float* __restrict__ tau,
                              const int* __restrict__ rows,
                              const int* __restrict__ cols,
                              float* __restrict__ Wt) {
    int e = blockIdx.x * blockDim.x + threadIdx.x;
    if (e >= N_EDGES) return;
    float wv = weights[e];
    int n = rows[e];
    int k = cols[e];
    float* base = Wt + (size_t)n * (N_RECEPT * N_BKG) + k;  // col = n*5+r, stride 100 per r
    #pragma unroll
    for (int r = 0; r < N_RECEPT; ++r) {
        atomicAdd(base + r * N_BKG, wv * tau[e * N_RECEPT + r]);
    }
}

// ---------------------------------------------------------------------------
// Kernel 3: fp32 WMMA GEMM.
//   C[t, col] = sum_k A[t, k] * B[k, col],  A = padded spikes [256 x 100],
//   B[k, col] = Wt[col*100 + k].
// One wave per 16x16 C tile; V_WMMA_F32_16X16X4_F32, K tiled by 4 (25 steps).
//
// Fragment layouts (wave32, per CDNA5 ISA 7.12.2):
//   A 16x4 f32: lane L (half=L/16, m=L%16): VGPR0=A[m, k0+2*half], VGPR1=A[m, k0+2*half+1]
//   B 4x16 f32: lane L: VGPR v -> B[k0 + v + 2*half, col0 + L%16]
//   C 16x16 f32: VGPR v -> row m0 + v + 8*half, col col0 + L%16
// ---------------------------------------------------------------------------
__global__ void __launch_bounds__(256)
wmma_gemm(const float* __restrict__ ws, float* __restrict__ out) {
    const float* __restrict__ Wt   = ws;
    const float* __restrict__ Apad = ws + SPAD_OFF;

    int wave = threadIdx.x >> 5;
    int lane = threadIdx.x & 31;
    int tile = blockIdx.x * 8 + wave;          // 0 .. 249999
    int ctile = tile % (N_COLS / 16);          // 0 .. 15624
    int mtile = tile / (N_COLS / 16);          // 0 .. 15

    int col0 = ctile * 16;
    int m0   = mtile * 16;
    int half = lane >> 4;                      // 0 or 1
    int l16  = lane & 15;

    // Per-lane base pointers for the k-loop (8-byte aligned: offsets are even)
    const float* aPtr = Apad + (size_t)(m0 + l16) * N_BKG + 2 * half;
    const float* bPtr = Wt   + (size_t)(col0 + l16) * N_BKG + 2 * half;

    v8f c = {};
    #pragma unroll
    for (int k0 = 0; k0 < N_BKG; k0 += 4) {
        v2f a = *(const v2f*)(aPtr + k0);
        v2f b = *(const v2f*)(bPtr + k0);
        // D = A x B + C   (fp32 WMMA, 16x16x4)
        c = __builtin_amdgcn_wmma_f32_16x16x4_f32(
                /*neg_a=*/false, a, /*neg_b=*/false, b,
                /*c_mod=*/(short)0, c, /*reuse_a=*/false, /*reuse_b=*/false);
    }

    int col = col0 + l16;
    #pragma unroll
    for (int v = 0; v < 8; ++v) {
        int m = m0 + v + 8 * half;
        if (m < SEQ_T) {
            out[(size_t)m * N_COLS + col] = c[v];
        }
    }
}

// ---------------------------------------------------------------------------
extern "C" void kernel_launch(void* const* d_in, const int* in_sizes, int n_in,
                              void* d_out, int out_size, void* d_ws, size_t ws_size,
                              hipStream_t stream) {
    const float* weights = (const float*)d_in[0];   // [500000]
    const float* tau_syn = (const float*)d_in[1];   // [500000, 5]
    const float* spikes  = (const float*)d_in[2];   // [250, 100]
    const int*   rows    = (const int*)d_in[3];     // [500000]
    const int*   cols    = (const int*)d_in[4];     // [500000]
    float* out = (float*)d_out;                     // [250 * 250000]
    float* ws  = (float*)d_ws;                      // needs ~100.1 MB

    (void)in_sizes; (void)n_in; (void)out_size; (void)ws_size;

    // 1) zero Wt + build padded spikes
    init_ws<<<2048, 256, 0, stream>>>(ws, spikes);

    // 2) scatter edges into dense Wt
    scatter_edges<<<(N_EDGES + 255) / 256, 256, 0, stream>>>(
        weights, tau_syn, rows, cols, ws);

    // 3) WMMA GEMM: 250000 tiles, 8 waves/block -> 31250 blocks
    wmma_gemm<<<(16 * (N_COLS / 16)) / 8, 256, 0, stream>>>(ws, out);
}